// SAGE_1211180778042
// MI455X (gfx1250) — compile-verified
//
#include <hip/hip_runtime.h>
#include <cstdint>
#include <cstddef>

// ---------------------------------------------------------------------------
// GraphSAGE (pool aggregator), 3 layers, for MI455X (gfx1250, wave32).
//   layer: m = relu(h@Wp+bp); agg = segment_max(m[src], dst) (>=0, empty->0);
//          h' = relu(h@Ws + agg@Wn + bn)
//   out  : log_softmax(h3)
// GEMMs use V_WMMA_F32_16X16X4_F32 (f32 exactness; GEMM cost ~ edge-traffic
// cost at the roofline, so no reason to drop precision).
// Scatter-max uses u32 atomicMax on float bits (valid since relu output >= 0).
// ---------------------------------------------------------------------------

typedef float v2f __attribute__((ext_vector_type(2)));
typedef float v8f __attribute__((ext_vector_type(8)));

#define NNODES 100000
#define NEDGES 800000
#define DIN    128          // every GEMM in this net has K = 128
#define SP     132          // LDS row stride (pad: 132 % 64 = 4 -> no bank conflicts)

// ------------------------------- zero fill ---------------------------------
__global__ void zero_f32(float* __restrict__ p, int n4) {
    int i = blockIdx.x * blockDim.x + threadIdx.x;
    if (i < n4) {
        float4 z = make_float4(0.f, 0.f, 0.f, 0.f);
        ((float4*)p)[i] = z;
    }
}

// ------------------------------- WMMA GEMM ---------------------------------
// out[row, col] = relu( sum_k A[row,k]*W[k,col] (+ A2@W2) + bias[col] )
// grid.x = NNODES/16 row panels; blockDim = 32 * (dout/16); wave w owns the
// 16x16 tile at columns [16w, 16w+16). A-tile staged via LDS; W from global
// (tiny, L2-resident). No partial tiles -> EXEC all ones for WMMA.
__global__ void __launch_bounds__(256)
sage_gemm(const float* __restrict__ A,  const float* __restrict__ W,
          const float* __restrict__ A2, const float* __restrict__ W2,
          const float* __restrict__ bias, float* __restrict__ out, int dout)
{
    __shared__ float Alds[16 * SP];

    const int tid      = threadIdx.x;
    const int lane     = tid & 31;
    const int wv       = tid >> 5;
    const int row_base = blockIdx.x * 16;
    const int n        = lane & 15;     // fragment row (A) / column (B,C)
    const int kh       = lane >> 4;     // K-half select within a K=4 step
    const int col      = wv * 16 + n;

    v8f c = {0.f, 0.f, 0.f, 0.f, 0.f, 0.f, 0.f, 0.f};

    const int npass = (A2 != nullptr) ? 2 : 1;
    for (int p = 0; p < npass; ++p) {
        const float* Ap = p ? A2 : A;
        const float* Wp = p ? W2 : W;

        if (p) __syncthreads();               // all reads of prev tile done
        // stage 16 x 128 A-tile (512 float4s, cooperative)
        for (int i = tid; i < 16 * (DIN / 4); i += blockDim.x) {
            int r  = i >> 5;                  // DIN/4 == 32
            int c4 = i & 31;
            float4 v = ((const float4*)(Ap + (size_t)(row_base + r) * DIN))[c4];
            *(float4*)&Alds[r * SP + c4 * 4] = v;
        }
        __syncthreads();

        // K loop: 32 steps of V_WMMA_F32_16X16X4_F32
        for (int k4 = 0; k4 < DIN; k4 += 4) {
            const int k0 = k4 + 2 * kh;
            // A frag 16x4: vgpr0={K=0|K=2}, vgpr1={K=1|K=3} across lane halves
            v2f a = *(const v2f*)&Alds[n * SP + k0];
            // B frag 4x16: same K striping, N across lanes 0-15
            v2f b;
            b.x = Wp[(size_t)k0 * dout + col];
            b.y = Wp[(size_t)(k0 + 1) * dout + col];
            c = __builtin_amdgcn_wmma_f32_16x16x4_f32(
                    false, a, false, b, (short)0, c, false, false);
        }
    }

    // epilogue: bias + relu, C layout: vgpr i -> row (i + 8*kh), col = col
    const float bv = bias[col];
    #pragma unroll
    for (int i = 0; i < 8; ++i) {
        int   row = row_base + i + 8 * kh;
        float v   = fmaxf(c[i] + bv, 0.f);
        out[(size_t)row * dout + col] = v;
    }
}

// --------------------------- edge scatter-max ------------------------------
// One wave per edge: lane grabs float4 of m[src], u32-atomicMax into agg[dst].
// Valid because m = relu(..) >= 0 (non-negative IEEE floats order as u32).
__global__ void __launch_bounds__(256)
edge_scatter_max(const float* __restrict__ m, const int* __restrict__ src,
                 const int* __restrict__ dst, unsigned int* __restrict__ agg,
                 int nE)
{
    int e = blockIdx.x * 8 + (threadIdx.x >> 5);
    if (e >= nE) return;
    int lane = threadIdx.x & 31;
    int s = src[e];
    int t = dst[e];
    float4 v = ((const float4*)(m + (size_t)s * DIN))[lane];
    unsigned int* ap = agg + (size_t)t * DIN + lane * 4;
    atomicMax(ap + 0, __float_as_uint(v.x));
    atomicMax(ap + 1, __float_as_uint(v.y));
    atomicMax(ap + 2, __float_as_uint(v.z));
    atomicMax(ap + 3, __float_as_uint(v.w));
}

// ------------------------------ log_softmax --------------------------------
// 64 classes: one wave per row, 2 values per lane, shfl_xor reductions.
__global__ void __launch_bounds__(256)
log_softmax64(float* __restrict__ out, int nRows)
{
    int r = blockIdx.x * 8 + (threadIdx.x >> 5);
    if (r >= nRows) return;
    int lane = threadIdx.x & 31;
    float* row = out + (size_t)r * 64;
    float v0 = row[lane];
    float v1 = row[lane + 32];
    float mx = fmaxf(v0, v1);
    #pragma unroll
    for (int off = 16; off; off >>= 1) mx = fmaxf(mx, __shfl_xor(mx, off, 32));
    float s = expf(v0 - mx) + expf(v1 - mx);
    #pragma unroll
    for (int off = 16; off; off >>= 1) s += __shfl_xor(s, off, 32);
    float l = mx + logf(s);
    row[lane]      = v0 - l;
    row[lane + 32] = v1 - l;
}

// ------------------------------- launcher ----------------------------------
extern "C" void kernel_launch(void* const* d_in, const int* in_sizes, int n_in,
                              void* d_out, int out_size, void* d_ws, size_t ws_size,
                              hipStream_t stream)
{
    (void)in_sizes; (void)n_in; (void)out_size; (void)ws_size;

    const float* x    = (const float*)d_in[0];
    const int*   esrc = (const int*)d_in[1];
    const int*   edst = (const int*)d_in[2];
    const float* Wp[3] = {(const float*)d_in[3],  (const float*)d_in[8],  (const float*)d_in[13]};
    const float* bp[3] = {(const float*)d_in[4],  (const float*)d_in[9],  (const float*)d_in[14]};
    const float* Ws[3] = {(const float*)d_in[5],  (const float*)d_in[10], (const float*)d_in[15]};
    const float* Wn[3] = {(const float*)d_in[6],  (const float*)d_in[11], (const float*)d_in[16]};
    const float* bn[3] = {(const float*)d_in[7],  (const float*)d_in[12], (const float*)d_in[17]};

    const size_t NB = (size_t)NNODES * DIN;            // floats per node buffer
    float* buf0 = (float*)d_ws;
    float* buf1 = buf0 + NB;
    float* buf2 = buf1 + NB;
    float* o    = (float*)d_out;

    const dim3 g(NNODES / 16);                         // 6250 row panels
    const int  n4    = (int)(NB / 4);                  // 3.2M float4
    const int  zgrid = (n4 + 255) / 256;
    const int  egrid = (NEDGES + 7) / 8;               // 1 wave / edge

    // ---- layer 0: h = x -> buf2 ----
    sage_gemm<<<g, 256, 0, stream>>>(x, Wp[0], nullptr, nullptr, bp[0], buf0, 128);
    zero_f32<<<zgrid, 256, 0, stream>>>(buf1, n4);
    edge_scatter_max<<<egrid, 256, 0, stream>>>(buf0, esrc, edst, (unsigned int*)buf1, NEDGES);
    sage_gemm<<<g, 256, 0, stream>>>(x, Ws[0], buf1, Wn[0], bn[0], buf2, 128);

    // ---- layer 1: h = buf2 -> buf0 ----
    sage_gemm<<<g, 256, 0, stream>>>(buf2, Wp[1], nullptr, nullptr, bp[1], buf0, 128);
    zero_f32<<<zgrid, 256, 0, stream>>>(buf1, n4);
    edge_scatter_max<<<egrid, 256, 0, stream>>>(buf0, esrc, edst, (unsigned int*)buf1, NEDGES);
    sage_gemm<<<g, 256, 0, stream>>>(buf2, Ws[1], buf1, Wn[1], bn[1], buf0, 128);

    // ---- layer 2: h = buf0 -> d_out (dout = 64) ----
    sage_gemm<<<g, 256, 0, stream>>>(buf0, Wp[2], nullptr, nullptr, bp[2], buf1, 128);
    zero_f32<<<zgrid, 256, 0, stream>>>(buf2, n4);
    edge_scatter_max<<<egrid, 256, 0, stream>>>(buf1, esrc, edst, (unsigned int*)buf2, NEDGES);
    sage_gemm<<<g, 128, 0, stream>>>(buf0, Ws[2], buf2, Wn[2], bn[2], o, 64);

    // ---- log_softmax over 64 classes, in place on d_out ----
    log_softmax64<<<(NNODES + 7) / 8, 256, 0, stream>>>(o, NNODES);
}